// TreeNNs_3204045603892
// MI455X (gfx1250) — compile-verified
//
#include <hip/hip_runtime.h>
#include <hip/hip_bf16.h>

#define B_ 2048
#define D_ 512
#define H_ 256
#define L_ 64
#define C_ 256

typedef __attribute__((ext_vector_type(2))) float v2f;
typedef __attribute__((ext_vector_type(8))) float v8f;

// ---------------------------------------------------------------------------
// Kernel 0: zero the per-leaf histogram
// ---------------------------------------------------------------------------
__global__ void k_init(int* counts) {
    if (threadIdx.x < L_) counts[threadIdx.x] = 0;
}

// ---------------------------------------------------------------------------
// Kernel 1: feat = relu(x @ Wd1 + bd1)   [2048x512] @ [512x256] -> [2048x256]
// Block: 128 threads (4 waves), 64x64 output tile.
// Both A (x) and B (Wd1) tiles staged in LDS via float4 (global_load_b128);
// WMMA operands read from LDS. Strides 68 (=4*17) keep 16B alignment and
// spread the 16-lane WMMA reads across banks.
// ---------------------------------------------------------------------------
__global__ __launch_bounds__(128) void k_feat(const float* __restrict__ x,
                                              const float* __restrict__ Wd1,
                                              const float* __restrict__ bd1,
                                              float* __restrict__ feat) {
    __shared__ float As[64 * 68];
    __shared__ float Bs[64 * 68];
    const int bn = blockIdx.x;            // 0..3   (N / 64)
    const int bm = blockIdx.y;            // 0..31  (M / 64)
    const int tid  = threadIdx.x;
    const int wave = tid >> 5;
    const int lane = tid & 31;
    const int ln   = lane & 15;
    const int koff = (lane >> 4) << 1;    // upper half-wave holds K+2,K+3
    const int rowBase = bm * 64;
    const int colBase = bn * 64;

    v8f acc[4];
#pragma unroll
    for (int i = 0; i < 4; ++i) acc[i] = (v8f)0.f;

    for (int k0 = 0; k0 < D_; k0 += 64) {
        __syncthreads();
        // stage x[rowBase..+64][k0..+64] and Wd1[k0..+64][colBase..+64] (b128)
#pragma unroll
        for (int e = tid; e < 1024; e += 128) {
            int r = e >> 4, c4 = (e & 15) << 2;
            *(float4*)&As[r * 68 + c4] =
                *(const float4*)&x[(rowBase + r) * D_ + k0 + c4];
            *(float4*)&Bs[r * 68 + c4] =
                *(const float4*)&Wd1[(k0 + r) * H_ + colBase + c4];
        }
        __syncthreads();
#pragma unroll 4
        for (int kk = 0; kk < 64; kk += 4) {
            const int arow = wave * 16 + ln;
            const int kb   = kk + koff;
            v2f a;
            a.x = As[arow * 68 + kb];
            a.y = As[arow * 68 + kb + 1];
#pragma unroll
            for (int nt = 0; nt < 4; ++nt) {
                const int n = nt * 16 + ln;
                v2f b;
                b.x = Bs[kb * 68 + n];
                b.y = Bs[(kb + 1) * 68 + n];
                acc[nt] = __builtin_amdgcn_wmma_f32_16x16x4_f32(
                    false, a, false, b, (short)0, acc[nt], false, false);
            }
        }
    }
    // C/D layout: VGPR j -> M=j (lanes 0-15) / M=j+8 (lanes 16-31); N = lane&15
    const int mhalf = (lane >> 4) * 8;
#pragma unroll
    for (int nt = 0; nt < 4; ++nt) {
        const int col  = colBase + nt * 16 + ln;
        const float bias = bd1[col];
#pragma unroll
        for (int j = 0; j < 8; ++j) {
            const int row = rowBase + wave * 16 + mhalf + j;
            float v = acc[nt][j] + bias;
            feat[row * H_ + col] = v > 0.f ? v : 0.f;
        }
    }
}

// ---------------------------------------------------------------------------
// Kernel 2: routing. argmax(softmax(z)) == argmax(z), softmax skipped.
// One block (64 threads) per sample, one leaf per thread; LDS argmax with
// first-index tie-break (jnp.argmax semantics); atomic histogram.
// ---------------------------------------------------------------------------
__global__ __launch_bounds__(64) void k_route(const float* __restrict__ feat,
                                              const float* __restrict__ Wd2,
                                              const float* __restrict__ bd2,
                                              int* __restrict__ choices,
                                              int* __restrict__ counts) {
    __shared__ float sval[L_];
    __shared__ int   sidx[L_];
    const int b = blockIdx.x;
    const int l = threadIdx.x;
    const float* f = feat + b * H_;
    float s = bd2[l];
#pragma unroll 4
    for (int k = 0; k < H_; ++k) s += f[k] * Wd2[k * L_ + l];
    sval[l] = s;
    sidx[l] = l;
    __syncthreads();
    for (int off = 32; off > 0; off >>= 1) {
        if (l < off) {
            float v2 = sval[l + off];
            int   i2 = sidx[l + off];
            if (v2 > sval[l] || (v2 == sval[l] && i2 < sidx[l])) {
                sval[l] = v2;
                sidx[l] = i2;
            }
        }
        __syncthreads();
    }
    if (l == 0) {
        choices[b] = sidx[0];
        atomicAdd(&counts[sidx[0]], 1);
    }
}

// ---------------------------------------------------------------------------
// Kernel 3: exclusive prefix scan over 64 leaf counts (trivial size)
// ---------------------------------------------------------------------------
__global__ void k_scan(const int* __restrict__ counts,
                       int* __restrict__ offsets,
                       int* __restrict__ cursor) {
    if (threadIdx.x == 0) {
        int acc = 0;
        for (int l = 0; l < L_; ++l) {
            offsets[l] = acc;
            cursor[l]  = acc;
            acc += counts[l];
        }
    }
}

// ---------------------------------------------------------------------------
// Kernel 4: scatter samples into per-leaf buckets
// ---------------------------------------------------------------------------
__global__ __launch_bounds__(256) void k_scatter(const int* __restrict__ choices,
                                                 int* __restrict__ cursor,
                                                 int* __restrict__ order) {
    const int b = blockIdx.x * 256 + threadIdx.x;
    if (b < B_) {
        const int l = choices[b];
        const int pos = atomicAdd(&cursor[l], 1);
        order[pos] = b;
    }
}

// ---------------------------------------------------------------------------
// Kernel 5: grouped expert GEMM.
// Grid (tile=128, leaf=64); block = 128 threads (4 waves); dead tiles exit.
// A tile (16x256 gathered feat rows, zero-padded) stays resident in LDS.
// Weight matrices are staged per 64-K chunk (64x256 f32, 66.6KB) via b128
// loads shared by all 4 waves; each wave owns 64 output columns.
//   Htile = relu(A @ W1[leaf] + b1[leaf]); preds = Htile @ W2[leaf] + b2[leaf]
// ---------------------------------------------------------------------------
__global__ __launch_bounds__(128) void k_expert(const float* __restrict__ feat,
                                                const float* __restrict__ W1,
                                                const float* __restrict__ b1,
                                                const float* __restrict__ W2,
                                                const float* __restrict__ b2,
                                                const int* __restrict__ counts,
                                                const int* __restrict__ offsets,
                                                const int* __restrict__ order,
                                                float* __restrict__ out) {
    __shared__ float As[16 * 260];
    __shared__ float Hs[16 * 260];
    __shared__ float Bs[64 * 260];
    const int leaf = blockIdx.y;
    const int t    = blockIdx.x;
    const int cnt  = counts[leaf];
    if (t * 16 >= cnt) return;
    const int base  = offsets[leaf] + t * 16;
    const int nrows = min(16, cnt - t * 16);

    const int tid   = threadIdx.x;
    const int wave  = tid >> 5;
    const int lane  = tid & 31;
    const int ln    = lane & 15;
    const int koff  = (lane >> 4) << 1;
    const int mhalf = (lane >> 4) * 8;

    // gather A tile (16 x 256) with b128 loads, zero-pad invalid rows
#pragma unroll
    for (int e = tid; e < 1024; e += 128) {
        int r = e >> 6, c4 = (e & 63) << 2;
        float4 v = make_float4(0.f, 0.f, 0.f, 0.f);
        if (r < nrows) v = *(const float4*)&feat[order[base + r] * H_ + c4];
        *(float4*)&As[r * 260 + c4] = v;
    }

    const float* W1p = W1 + (size_t)leaf * H_ * H_;
    const float* W2p = W2 + (size_t)leaf * H_ * C_;

    // ---- layer 1:  Hs = relu(A @ W1 + b1) ----
    v8f acc[4];
#pragma unroll
    for (int i = 0; i < 4; ++i) acc[i] = (v8f)0.f;
    for (int k0 = 0; k0 < H_; k0 += 64) {
        __syncthreads();
#pragma unroll
        for (int e = tid; e < 4096; e += 128) {    // 64x256 chunk of W1
            int r = e >> 6, c4 = (e & 63) << 2;
            *(float4*)&Bs[r * 260 + c4] =
                *(const float4*)&W1p[(k0 + r) * H_ + c4];
        }
        __syncthreads();
#pragma unroll 4
        for (int kk = 0; kk < 64; kk += 4) {
            const int kb = kk + koff;
            v2f a;
            a.x = As[ln * 260 + k0 + kb];
            a.y = As[ln * 260 + k0 + kb + 1];
#pragma unroll
            for (int nt = 0; nt < 4; ++nt) {
                const int n = wave * 64 + nt * 16 + ln;
                v2f bb;
                bb.x = Bs[kb * 260 + n];
                bb.y = Bs[(kb + 1) * 260 + n];
                acc[nt] = __builtin_amdgcn_wmma_f32_16x16x4_f32(
                    false, a, false, bb, (short)0, acc[nt], false, false);
            }
        }
    }
    __syncthreads();           // all WMMA reads of Bs done before Hs writes? (Hs distinct; barrier orders Hs writes vs layer-2 reads)
#pragma unroll
    for (int nt = 0; nt < 4; ++nt) {
        const int col  = wave * 64 + nt * 16 + ln;
        const float bias = b1[leaf * H_ + col];
#pragma unroll
        for (int j = 0; j < 8; ++j) {
            const int row = mhalf + j;
            float v = acc[nt][j] + bias;
            Hs[row * 260 + col] = v > 0.f ? v : 0.f;
        }
    }

    // ---- layer 2:  out = Hs @ W2 + b2 ----
#pragma unroll
    for (int i = 0; i < 4; ++i) acc[i] = (v8f)0.f;
    for (int k0 = 0; k0 < H_; k0 += 64) {
        __syncthreads();
#pragma unroll
        for (int e = tid; e < 4096; e += 128) {    // 64x256 chunk of W2
            int r = e >> 6, c4 = (e & 63) << 2;
            *(float4*)&Bs[r * 260 + c4] =
                *(const float4*)&W2p[(k0 + r) * C_ + c4];
        }
        __syncthreads();
#pragma unroll 4
        for (int kk = 0; kk < 64; kk += 4) {
            const int kb = kk + koff;
            v2f a;
            a.x = Hs[ln * 260 + k0 + kb];
            a.y = Hs[ln * 260 + k0 + kb + 1];
#pragma unroll
            for (int nt = 0; nt < 4; ++nt) {
                const int n = wave * 64 + nt * 16 + ln;
                v2f bb;
                bb.x = Bs[kb * 260 + n];
                bb.y = Bs[(kb + 1) * 260 + n];
                acc[nt] = __builtin_amdgcn_wmma_f32_16x16x4_f32(
                    false, a, false, bb, (short)0, acc[nt], false, false);
            }
        }
    }
    // bias + scatter rows to output
#pragma unroll
    for (int nt = 0; nt < 4; ++nt) {
        const int col  = wave * 64 + nt * 16 + ln;
        const float bias = b2[leaf * C_ + col];
#pragma unroll
        for (int j = 0; j < 8; ++j) {
            const int row = mhalf + j;
            if (row < nrows) {
                const int smp = order[base + row];
                out[smp * C_ + col] = acc[nt][j] + bias;
            }
        }
    }
}

// ---------------------------------------------------------------------------
extern "C" void kernel_launch(void* const* d_in, const int* in_sizes, int n_in,
                              void* d_out, int out_size, void* d_ws, size_t ws_size,
                              hipStream_t stream) {
    const float* x   = (const float*)d_in[0];
    const float* Wd1 = (const float*)d_in[1];
    const float* bd1 = (const float*)d_in[2];
    const float* Wd2 = (const float*)d_in[3];
    const float* bd2 = (const float*)d_in[4];
    const float* W1  = (const float*)d_in[5];
    const float* b1  = (const float*)d_in[6];
    const float* W2  = (const float*)d_in[7];
    const float* b2  = (const float*)d_in[8];
    float* out = (float*)d_out;

    // workspace layout
    float* feat   = (float*)d_ws;              // B*H floats (2 MB)
    int* choices  = (int*)(feat + B_ * H_);    // B
    int* counts   = choices + B_;              // L
    int* offsets  = counts + L_;               // L
    int* cursor   = offsets + L_;              // L
    int* order    = cursor + L_;               // B

    k_init<<<1, 64, 0, stream>>>(counts);
    k_feat<<<dim3(H_ / 64, B_ / 64), 128, 0, stream>>>(x, Wd1, bd1, feat);
    k_route<<<B_, 64, 0, stream>>>(feat, Wd2, bd2, choices, counts);
    k_scan<<<1, 32, 0, stream>>>(counts, offsets, cursor);
    k_scatter<<<B_ / 256, 256, 0, stream>>>(choices, cursor, order);
    k_expert<<<dim3(B_ / 16, L_), 128, 0, stream>>>(feat, W1, b1, W2, b2,
                                                    counts, offsets, order, out);
}